// SpikeLayer_62466004353543
// MI455X (gfx1250) — compile-verified
//
#include <hip/hip_runtime.h>

// One LIF neuron-update step (SpikeLayer.update_neurons), fp32 elementwise.
// Bandwidth-bound: 28 B/element, ~470 MB -> ~20 us at 23.3 TB/s HBM.
//
// Primary path (CDNA5 async engine): double-buffered GLOBAL_LOAD_ASYNC_TO_LDS_B128
// pipeline — issue next tile's 3 async b128 loads, s_wait_asynccnt 3, consume the
// current tile from LDS (ds_load_b128, lane-private slots => no barrier needed),
// store results straight from VGPRs as non-temporal b128 (output set > L2).
// Fallback path (if builtin unavailable): direct NT b128 loads + global_prefetch_b8.

typedef __attribute__((ext_vector_type(4))) float v4f;
typedef __attribute__((ext_vector_type(4))) int   v4i;

#if __has_builtin(__builtin_amdgcn_global_load_async_to_lds_b128)
  #define HAVE_ASYNC_LDS 1
#endif

#if __has_builtin(__builtin_amdgcn_s_wait_asynccnt)
  #define WAIT_ASYNC(imm) __builtin_amdgcn_s_wait_asynccnt(imm)
#else
  #define WAIT_ASYNC(imm) asm volatile("s_wait_asynccnt " #imm ::: "memory")
#endif

// Builtin signature (from clang diagnostic): v4i addrspace(1)* src,
// v4i addrspace(3)* dst, imm offset, imm cpol.
#define AS_GLOBAL_V4I(p) ((__attribute__((address_space(1))) v4i*)(p))
#define AS_LDS_V4I(p)    ((__attribute__((address_space(3))) v4i*)(p))

namespace {
constexpr float kTime      = 5.0f;  // simulation time
constexpr float kRefracNew = 7.0f;  // TIME + TAU_REFRAC
constexpr float kVth       = 1.0f;  // v_thresh
constexpr float kLeak      = 0.1f;  // 0.1 * dt
}  // namespace

__device__ __forceinline__ void lif4(const v4f imp, const v4f mem, const v4f ref,
                                     v4f& psp, v4f& memo, v4f& refo, v4f& st) {
#pragma unroll
  for (int k = 0; k < 4; ++k) {
    float m = mem[k] + ((ref[k] > kTime) ? 0.0f : imp[k]);  // refractory mask
    m = (m > 0.0f) ? (m - kLeak) : m;                       // leak on positive mem
    const bool s = (m >= kVth);                             // spike (post-leak)
    psp[k]  = s ? kVth : 0.0f;
    memo[k] = s ? 0.0f : m;
    refo[k] = s ? kRefracNew : ref[k];
    st[k]   = s ? kTime : 0.0f;
  }
}

__global__ __launch_bounds__(256)
void spike_lif_kernel(const float* __restrict__ g_imp,
                      const float* __restrict__ g_mem,
                      const float* __restrict__ g_ref,
                      float* __restrict__ g_psp,
                      float* __restrict__ g_memo,
                      float* __restrict__ g_refo,
                      float* __restrict__ g_sto,
                      int n) {
  const int nvec  = n >> 2;  // float4 count
  const int gsize = (int)(gridDim.x * blockDim.x);
  const int gtid  = (int)(blockIdx.x * blockDim.x + threadIdx.x);

  const v4f* __restrict__ imp4 = (const v4f*)g_imp;
  const v4f* __restrict__ mem4 = (const v4f*)g_mem;
  const v4f* __restrict__ ref4 = (const v4f*)g_ref;
  v4f* __restrict__ psp4  = (v4f*)g_psp;
  v4f* __restrict__ memo4 = (v4f*)g_memo;
  v4f* __restrict__ refo4 = (v4f*)g_refo;
  v4f* __restrict__ sto4  = (v4f*)g_sto;

#if defined(HAVE_ASYNC_LDS)
  // ---- CDNA5 async global->LDS double-buffered pipeline --------------------
  // Tile = 256 float4 per stream per block (one v4f per lane). LDS use:
  // 2 bufs * 3 streams * 256 * 16B = 24 KB (of 320 KB/WGP).
  __shared__ v4f s_imp[2][256];
  __shared__ v4f s_mem[2][256];
  __shared__ v4f s_ref[2][256];

  const int lane    = (int)threadIdx.x;
  const int ntiles  = nvec >> 8;          // full 256-vec tiles
  const int tstride = (int)gridDim.x;

  auto issue_tile = [&](int tile, int buf) {
    const int i = (tile << 8) + lane;     // always in range for full tiles
    __builtin_amdgcn_global_load_async_to_lds_b128(
        AS_GLOBAL_V4I(imp4 + i), AS_LDS_V4I(&s_imp[buf][lane]), 0, 0);
    __builtin_amdgcn_global_load_async_to_lds_b128(
        AS_GLOBAL_V4I(mem4 + i), AS_LDS_V4I(&s_mem[buf][lane]), 0, 0);
    __builtin_amdgcn_global_load_async_to_lds_b128(
        AS_GLOBAL_V4I(ref4 + i), AS_LDS_V4I(&s_ref[buf][lane]), 0, 0);
  };

  int tile = (int)blockIdx.x;
  int buf  = 0;
  if (tile < ntiles) issue_tile(tile, 0);  // prologue

  for (; tile < ntiles; tile += tstride) {
    const int  nxt  = tile + tstride;
    const bool more = (nxt < ntiles);
    if (more) issue_tile(nxt, buf ^ 1);    // keep 6 b128 reqs/lane in flight
    if (more) { WAIT_ASYNC(3); } else { WAIT_ASYNC(0); }
    asm volatile("" ::: "memory");         // don't hoist LDS reads above wait

    // Lane-private LDS slots: the async engine wrote exactly what this lane
    // reads, and ASYNCcnt orders it for the issuing wave -> no barrier.
    const v4f imp = s_imp[buf][lane];
    const v4f mem = s_mem[buf][lane];
    const v4f ref = s_ref[buf][lane];

    v4f psp, memo, refo, st;
    lif4(imp, mem, ref, psp, memo, refo, st);

    const int i = (tile << 8) + lane;
    __builtin_nontemporal_store(psp,  psp4  + i);
    __builtin_nontemporal_store(memo, memo4 + i);
    __builtin_nontemporal_store(refo, refo4 + i);
    __builtin_nontemporal_store(st,   sto4  + i);
    buf ^= 1;
  }
  const int vec_done = ntiles << 8;
#else
  const int vec_done = 0;
#endif

  // ---- Remainder vecs (fallback: the whole range) — direct NT b128 path ----
  for (int i = vec_done + gtid; i < nvec; i += gsize) {
    const int pf = i + gsize;
    if (pf < nvec) {  // global_prefetch_b8 one grid-stride iteration ahead
      __builtin_prefetch((const void*)(imp4 + pf), 0, 0);
      __builtin_prefetch((const void*)(mem4 + pf), 0, 0);
      __builtin_prefetch((const void*)(ref4 + pf), 0, 0);
    }
    const v4f imp = __builtin_nontemporal_load(imp4 + i);
    const v4f mem = __builtin_nontemporal_load(mem4 + i);
    const v4f ref = __builtin_nontemporal_load(ref4 + i);
    v4f psp, memo, refo, st;
    lif4(imp, mem, ref, psp, memo, refo, st);
    __builtin_nontemporal_store(psp,  psp4  + i);
    __builtin_nontemporal_store(memo, memo4 + i);
    __builtin_nontemporal_store(refo, refo4 + i);
    __builtin_nontemporal_store(st,   sto4  + i);
  }

  // ---- Scalar tail (n % 4 != 0; not hit for 64*256*32*32, but safe) --------
  const int tail_base = nvec << 2;
  const int tail = n - tail_base;
  if (gtid < tail) {
    const int i = tail_base + gtid;
    float m = g_mem[i] + ((g_ref[i] > kTime) ? 0.0f : g_imp[i]);
    m = (m > 0.0f) ? (m - kLeak) : m;
    const bool s = (m >= kVth);
    g_psp[i]  = s ? kVth : 0.0f;
    g_memo[i] = s ? 0.0f : m;
    g_refo[i] = s ? kRefracNew : g_ref[i];
    g_sto[i]  = s ? kTime : 0.0f;
  }
}

extern "C" void kernel_launch(void* const* d_in, const int* in_sizes, int n_in,
                              void* d_out, int out_size, void* d_ws, size_t ws_size,
                              hipStream_t stream) {
  // Inputs (setup_inputs order): impulse, mem, refrac_until, spiketrain(unused).
  const float* imp = (const float*)d_in[0];
  const float* mem = (const float*)d_in[1];
  const float* ref = (const float*)d_in[2];
  const int n = in_sizes[0];

  // Outputs concatenated flat in return order: psp, mem, refrac, spiketrain.
  float* psp  = (float*)d_out;
  float* memo = psp  + n;
  float* refo = memo + n;
  float* sto  = refo + n;

  // For SHAPE=64*256*32*32: nvec = 4,194,304 -> 16384 tiles; cap at 4096
  // blocks so each block pipelines >=4 tiles (async double-buffering pays off),
  // 4096 blocks * 8 wave32 = 32K waves still saturates every WGP.
  const int nvec = n / 4;
  int blocks = (nvec + 255) / 256;
  if (blocks > 4096) blocks = 4096;
  if (blocks < 1) blocks = 1;

  spike_lif_kernel<<<blocks, 256, 0, stream>>>(imp, mem, ref,
                                               psp, memo, refo, sto, n);
}